// RNN_45561013076391
// MI455X (gfx1250) — compile-verified
//
#include <hip/hip_runtime.h>

// ---------------------------------------------------------------------------
// LSTM(H=5, B=8192, T=512) + MLP head for MI455X / gfx1250 (wave32, WMMA).
//
// Phase 1 (batch-dense, WMMA): z[t][b][g] = sum_h x[b,t,h]*W_ih[g,h] + (b_ih+b_hh)[g]
//   using V_WMMA_F32_16X16X4_F32, 16 batch rows per wave-tile, K=5 in two
//   K=4 chunks, N=20 gates as a 16-gate tile + 4-gate tile. Stored [T,B,20]
//   in d_ws so phase 2 streams it fully coalesced.
// Phase 2 (latency-bound recurrence): 1 lane per sequence, W_hh in VGPRs,
//   v_tanh_f32 for all activations, global_prefetch_b8 look-ahead on z,
//   MLP head fused at the end with weights in LDS.
// ---------------------------------------------------------------------------

typedef __attribute__((ext_vector_type(2))) float v2f;
typedef __attribute__((ext_vector_type(4))) float v4f;
typedef __attribute__((ext_vector_type(8))) float v8f;

#define B_   8192
#define T_   512
#define H_   5
#define G_   20     // 4*H

static __device__ __forceinline__ v8f wmma16x16x4(v2f a, v2f b, v8f c) {
  // (neg_a, A, neg_b, B, c_mod, C, reuse_a, reuse_b)
  return __builtin_amdgcn_wmma_f32_16x16x4_f32(false, a, false, b, (short)0, c,
                                               false, false);
}

static __device__ __forceinline__ float fast_tanh(float x) {
#if __has_builtin(__builtin_amdgcn_tanhf)
  return __builtin_amdgcn_tanhf(x);
#else
  float r;
  asm volatile("v_tanh_f32 %0, %1" : "=v"(r) : "v"(x));
  return r;
#endif
}
static __device__ __forceinline__ float fast_sig(float x) {
  return fmaf(fast_tanh(x * 0.5f), 0.5f, 0.5f);
}

// ---------------------------------------------------------------------------
// Phase 1: input projection via WMMA. One wave per (16-batch x 1-timestep) tile.
// ---------------------------------------------------------------------------
__global__ void __launch_bounds__(256) xproj_wmma_kernel(
    const float* __restrict__ x, const float* __restrict__ Wih,
    const float* __restrict__ bih, const float* __restrict__ bhh,
    float* __restrict__ z)
{
  const int lane = threadIdx.x & 31;
  const int wave = threadIdx.x >> 5;
  const int tile = blockIdx.x * 8 + wave;      // 512*512 tiles total
  const int t    = tile & (T_ - 1);
  const int b0   = (tile >> 9) << 4;           // (tile / T_) * 16
  const int half = lane >> 4;                  // K-split half (ISA A/B layout)
  const int l16  = lane & 15;
  const int k0   = half ? 2 : 0;

  // A-matrix: W_ih rows (M = gate). Lane m holds row m; lanes16-31 hold K=2,3.
  v2f a1c0, a1c1, a2c0, a2c1;
  {
    const int m = l16;                         // gates 0..15
    a1c0.x = Wih[m * H_ + k0];
    a1c0.y = Wih[m * H_ + k0 + 1];
    a1c1.x = half ? 0.0f : Wih[m * H_ + 4];    // K=4 only; K=5..7 pad
    a1c1.y = 0.0f;
  }
  {
    const int g   = 16 + l16;                  // gates 16..31, valid < 20
    const bool ok = (g < G_);
    const int gi  = ok ? g : 0;
    a2c0.x = ok ? Wih[gi * H_ + k0]     : 0.0f;
    a2c0.y = ok ? Wih[gi * H_ + k0 + 1] : 0.0f;
    a2c1.x = (ok && !half) ? Wih[gi * H_ + 4] : 0.0f;
    a2c1.y = 0.0f;
  }

  // B-matrix: x columns (N = batch). Shared by both gate tiles.
  const float* xr = x + ((size_t)(b0 + l16) * T_ + t) * H_;
  v2f bm0, bm1;
  bm0.x = xr[k0];
  bm0.y = xr[k0 + 1];
  bm1.x = half ? 0.0f : xr[4];
  bm1.y = 0.0f;

  // C init: fused bias broadcast along N. VGPR r <-> gate (r + 8*half).
  v8f c1, c2;
#pragma unroll
  for (int r = 0; r < 8; ++r) {
    const int g1 = r + 8 * half;
    const int g2 = 16 + r + 8 * half;
    c1[r] = bih[g1] + bhh[g1];
    c2[r] = (g2 < G_) ? (bih[g2] + bhh[g2]) : 0.0f;
  }

  c1 = wmma16x16x4(a1c0, bm0, c1);
  c1 = wmma16x16x4(a1c1, bm1, c1);
  c2 = wmma16x16x4(a2c0, bm0, c2);
  c2 = wmma16x16x4(a2c1, bm1, c2);

  // Store z[t][b0+n][0..19]; D rows are contiguous gates -> b128 stores.
  float* zr = z + ((size_t)t * B_ + b0 + l16) * G_;
  v4f lo = {c1[0], c1[1], c1[2], c1[3]};
  v4f hi = {c1[4], c1[5], c1[6], c1[7]};
  if (half == 0) {
    *(v4f*)(zr + 0)  = lo;                      // gates 0..7
    *(v4f*)(zr + 4)  = hi;
    v4f t2 = {c2[0], c2[1], c2[2], c2[3]};      // gates 16..19
    *(v4f*)(zr + 16) = t2;
  } else {
    *(v4f*)(zr + 8)  = lo;                      // gates 8..15
    *(v4f*)(zr + 12) = hi;
  }
}

// ---------------------------------------------------------------------------
// Phase 2: recurrence (1 lane per sequence) + fused MLP head.
// ---------------------------------------------------------------------------
__global__ void __launch_bounds__(64) lstm_rec_head_kernel(
    const float* __restrict__ x,  const float* __restrict__ Whh,
    const float* __restrict__ W1, const float* __restrict__ b1,
    const float* __restrict__ W2, const float* __restrict__ b2,
    const float* __restrict__ W3, const float* __restrict__ b3,
    const float* __restrict__ z,  float* __restrict__ out)
{
  __shared__ float sW1[32 * H_], sb1[32], sW2[32 * 32], sb2[32],
                   sW3[H_ * 32], sb3[H_];
  for (int i = threadIdx.x; i < 32 * H_;  i += blockDim.x) sW1[i] = W1[i];
  for (int i = threadIdx.x; i < 32;       i += blockDim.x) sb1[i] = b1[i];
  for (int i = threadIdx.x; i < 32 * 32;  i += blockDim.x) sW2[i] = W2[i];
  for (int i = threadIdx.x; i < 32;       i += blockDim.x) sb2[i] = b2[i];
  for (int i = threadIdx.x; i < H_ * 32;  i += blockDim.x) sW3[i] = W3[i];
  for (int i = threadIdx.x; i < H_;       i += blockDim.x) sb3[i] = b3[i];
  __syncthreads();

  const int b = blockIdx.x * blockDim.x + threadIdx.x;

  // Recurrent weights live in registers (100 floats/lane).
  float w[G_][H_];
#pragma unroll
  for (int g = 0; g < G_; ++g)
#pragma unroll
    for (int k = 0; k < H_; ++k) w[g][k] = Whh[g * H_ + k];

  float h[H_] = {0.f, 0.f, 0.f, 0.f, 0.f};
  float c[H_] = {0.f, 0.f, 0.f, 0.f, 0.f};

  const float* zb = z + (size_t)b * G_;
  const size_t zstride = (size_t)B_ * G_;

  for (int t = 0; t < T_; ++t) {
    const float* zp = zb + (size_t)t * zstride;
    if (t + 8 < T_) __builtin_prefetch(zp + 8 * zstride, 0, 1);

    v4f z0 = *(const v4f*)(zp + 0);
    v4f z1 = *(const v4f*)(zp + 4);
    v4f z2 = *(const v4f*)(zp + 8);
    v4f z3 = *(const v4f*)(zp + 12);
    v4f z4 = *(const v4f*)(zp + 16);
    float a[G_] = {z0.x, z0.y, z0.z, z0.w, z1.x, z1.y, z1.z, z1.w,
                   z2.x, z2.y, z2.z, z2.w, z3.x, z3.y, z3.z, z3.w,
                   z4.x, z4.y, z4.z, z4.w};

#pragma unroll
    for (int g = 0; g < G_; ++g) {
      float s = a[g];
#pragma unroll
      for (int k = 0; k < H_; ++k) s = fmaf(w[g][k], h[k], s);
      a[g] = s;
    }

#pragma unroll
    for (int j = 0; j < H_; ++j) {
      const float ig = fast_sig(a[j]);
      const float fg = fast_sig(a[H_ + j]);
      const float gg = fast_tanh(a[2 * H_ + j]);
      const float og = fast_sig(a[3 * H_ + j]);
      c[j] = fmaf(fg, c[j], ig * gg);
      h[j] = og * fast_tanh(c[j]);
    }
  }

  // Residual + MLP head (weights broadcast from LDS).
  float in5[H_];
  const float* xl = x + ((size_t)b * T_ + (T_ - 1)) * H_;
#pragma unroll
  for (int j = 0; j < H_; ++j) in5[j] = h[j] + xl[j];

  float a1[32];
#pragma unroll
  for (int o = 0; o < 32; ++o) {
    float s = sb1[o];
#pragma unroll
    for (int k = 0; k < H_; ++k) s = fmaf(sW1[o * H_ + k], in5[k], s);
    a1[o] = fmaxf(s, 0.0f);
  }
  float a2[32];
#pragma unroll
  for (int o = 0; o < 32; ++o) {
    float s = sb2[o];
#pragma unroll
    for (int k = 0; k < 32; k += 4) {
      v4f wv = *(const v4f*)(sW2 + o * 32 + k);
      s = fmaf(wv.x, a1[k], s);
      s = fmaf(wv.y, a1[k + 1], s);
      s = fmaf(wv.z, a1[k + 2], s);
      s = fmaf(wv.w, a1[k + 3], s);
    }
    a2[o] = fmaxf(s, 0.0f);
  }
#pragma unroll
  for (int o = 0; o < H_; ++o) {
    float s = sb3[o];
#pragma unroll
    for (int k = 0; k < 32; ++k) s = fmaf(sW3[o * 32 + k], a2[k], s);
    out[(size_t)b * H_ + o] = s;
  }
}

// ---------------------------------------------------------------------------
extern "C" void kernel_launch(void* const* d_in, const int* in_sizes, int n_in,
                              void* d_out, int out_size, void* d_ws, size_t ws_size,
                              hipStream_t stream) {
  const float* x   = (const float*)d_in[0];
  const float* Wih = (const float*)d_in[1];
  const float* Whh = (const float*)d_in[2];
  const float* bih = (const float*)d_in[3];
  const float* bhh = (const float*)d_in[4];
  const float* W1  = (const float*)d_in[5];
  const float* b1  = (const float*)d_in[6];
  const float* W2  = (const float*)d_in[7];
  const float* b2  = (const float*)d_in[8];
  const float* W3  = (const float*)d_in[9];
  const float* b3  = (const float*)d_in[10];
  float* out = (float*)d_out;
  float* z   = (float*)d_ws;   // [T_, B_, 20] f32 = 335,544,320 bytes

  // Phase 1: (B/16)*T = 262144 wave-tiles, 8 waves (256 thr) per block.
  dim3 blk1(256), grd1((B_ / 16) * T_ / 8);
  hipLaunchKernelGGL(xproj_wmma_kernel, grd1, blk1, 0, stream,
                     x, Wih, bih, bhh, z);

  // Phase 2: one lane per sequence, small blocks to spread across WGPs.
  dim3 blk2(64), grd2(B_ / 64);
  hipLaunchKernelGGL(lstm_rec_head_kernel, grd2, blk2, 0, stream,
                     x, Whh, W1, b1, W2, b2, W3, b3, z, out);
}